// BiDecoderRNN_55825984913642
// MI455X (gfx1250) — compile-verified
//
#include <hip/hip_runtime.h>
#include <hip/hip_bf16.h>

typedef __attribute__((ext_vector_type(2))) float v2f;
typedef __attribute__((ext_vector_type(8))) float v8f;

#define HIDDEN 256
#define ATTN_F 100
#define GLOVE 50
#define VOCAB 50000
#define VOCAB_F 100
#define L_ENC 400
#define T_DEC 50
#define BATCH 16
#define BIG_V 50050

// workspace offsets (floats)
#define WS_H0   0
#define WS_C0   4096
#define WS_XEMB 8192
#define WS_ODEC 48192
#define WS_P    252992
#define WS_PV1  892992
#define WS_PGEN 972992
#define WS_AW   973792
#define WS_RMAX 1293792
#define WS_RSUM 1294592

// output offsets (floats)
#define OUT_PV  0
#define OUT_CL  40040000
#define OUT_COV 40040800
#define OUT_HT  40047200
#define OUT_CT  40051296

__device__ __forceinline__ float sigf(float x) { return 1.f / (1.f + __expf(-x)); }

// ---- wave-level 16x16 f32 WMMA tile GEMM, K-loop in steps of 4 -------------
// A: 16(M) x K, B: K x 16(N).  Per ISA 7.12.2:
//   A frag: lane l, vgpr v -> (m = l&15, k = 2*(l>>4)+v)
//   B frag: lane l, vgpr v -> (k = 2*(l>>4)+v, n = l&15)
//   D:      lane l, elem v -> (m = v + 8*(l>>4), n = l&15)
template <typename FA, typename FB>
__device__ __forceinline__ v8f wmma_gemm16(int lane, int K, FA loadA, FB loadB, v8f acc) {
  int idx = lane & 15;
  int kh  = (lane >> 4) << 1;  // 0 or 2
  for (int k0 = 0; k0 < K; k0 += 4) {
    v2f a, b;
    a.x = loadA(idx, k0 + kh);
    a.y = loadA(idx, k0 + kh + 1);
    b.x = loadB(k0 + kh, idx);
    b.y = loadB(k0 + kh + 1, idx);
    acc = __builtin_amdgcn_wmma_f32_16x16x4_f32(false, a, false, b, (short)0, acc,
                                                false, false);
  }
  return acc;
}

template <typename FS>
__device__ __forceinline__ void wmma_store16(int lane, const v8f& acc, FS store) {
  int n  = lane & 15;
  int mh = (lane >> 4) << 3;  // 0 or 8
#pragma unroll
  for (int r = 0; r < 8; ++r) store(mh + r, n, acc[r]);
}

// ---- K0: h0/c0 = concat(state[0],state[1]) @ reduce_W.T + reduce_b ---------
__global__ __launch_bounds__(1024) void k_reduce_init(
    const float* __restrict__ hidden_enc, const float* __restrict__ cell_enc,
    const float* __restrict__ reduce_W, const float* __restrict__ reduce_b,
    float* __restrict__ h0, float* __restrict__ c0) {
  int tid = threadIdx.x, lane = tid & 31, wave = tid >> 5;
  const float* src = (wave < 16) ? hidden_enc : cell_enc;
  float* dst = (wave < 16) ? h0 : c0;
  int nt = wave & 15;
  v8f acc = {};
  acc = wmma_gemm16(lane, 2 * HIDDEN,
      [&](int m, int k) {
        return src[(k < HIDDEN ? 0 : BATCH * HIDDEN) + m * HIDDEN + (k & (HIDDEN - 1))];
      },
      [&](int k, int n) { return reduce_W[(nt * 16 + n) * 2 * HIDDEN + k]; }, acc);
  wmma_store16(lane, acc, [&](int m, int n, float v) {
    int c = nt * 16 + n;
    dst[m * HIDDEN + c] = v + reduce_b[c];
  });
}

// ---- K1: embedding gather --------------------------------------------------
__global__ void k_embed(const float* __restrict__ table, const int* __restrict__ input_dec,
                        float* __restrict__ xemb) {
  int i = blockIdx.x * blockDim.x + threadIdx.x;
  if (i >= T_DEC * BATCH * GLOVE) return;
  int g = i % GLOVE, tb = i / GLOVE;
  xemb[i] = table[input_dec[tb] * GLOVE + g];
}

// ---- K2: persistent-block LSTM scan ----------------------------------------
__global__ __launch_bounds__(512) void k_lstm(
    const float* __restrict__ W_ih, const float* __restrict__ W_hh,
    const float* __restrict__ b_ih, const float* __restrict__ b_hh,
    const float* __restrict__ h0, const float* __restrict__ c0,
    const float* __restrict__ xemb, float* __restrict__ odec,
    float* __restrict__ hT, float* __restrict__ cT) {
  __shared__ float s_h[BATCH * HIDDEN];
  __shared__ float s_c[BATCH * HIDDEN];
  __shared__ float s_g[BATCH * 4 * HIDDEN];
  __shared__ float s_x[BATCH * GLOVE];
  int tid = threadIdx.x, lane = tid & 31, wave = tid >> 5;
  for (int i = tid; i < BATCH * HIDDEN; i += 512) { s_h[i] = h0[i]; s_c[i] = c0[i]; }
  __syncthreads();
  for (int t = 0; t < T_DEC; ++t) {
    for (int i = tid; i < BATCH * GLOVE; i += 512) s_x[i] = xemb[t * BATCH * GLOVE + i];
    __syncthreads();
    for (int nt = wave; nt < 64; nt += 16) {
      int col0 = nt * 16;
      v8f acc = {};
      acc = wmma_gemm16(lane, 52,  // K=50 padded to 52
          [&](int m, int k) { return k < GLOVE ? s_x[m * GLOVE + k] : 0.f; },
          [&](int k, int n) { return k < GLOVE ? W_ih[(col0 + n) * GLOVE + k] : 0.f; }, acc);
      acc = wmma_gemm16(lane, HIDDEN,
          [&](int m, int k) { return s_h[m * HIDDEN + k]; },
          [&](int k, int n) { return W_hh[(col0 + n) * HIDDEN + k]; }, acc);
      wmma_store16(lane, acc, [&](int m, int n, float v) {
        int c = col0 + n;
        s_g[m * 4 * HIDDEN + c] = v + b_ih[c] + b_hh[c];
      });
    }
    __syncthreads();
    for (int i = tid; i < BATCH * HIDDEN; i += 512) {
      int b = i >> 8, j = i & 255;
      const float* g = s_g + b * 4 * HIDDEN;
      float gi = g[j], gf = g[HIDDEN + j], gg = g[2 * HIDDEN + j], go = g[3 * HIDDEN + j];
      float c = sigf(gf) * s_c[i] + sigf(gi) * tanhf(gg);
      float h = sigf(go) * tanhf(c);
      s_c[i] = c; s_h[i] = h;
      odec[t * BATCH * HIDDEN + i] = h;
    }
    __syncthreads();
  }
  for (int i = tid; i < BATCH * HIDDEN; i += 512) { hT[i] = s_h[i]; cT[i] = s_c[i]; }
}

// ---- K3: time-invariant attention precompute P = enc @ Wenc.T + attn_b -----
__global__ __launch_bounds__(256) void k_attn_pre(
    const float* __restrict__ output_enc, const float* __restrict__ attn_W,
    const float* __restrict__ attn_b, float* __restrict__ P) {
  int tid = threadIdx.x, lane = tid & 31, wave = tid >> 5;
  int wid = blockIdx.x * 8 + wave;
  if (wid >= 400 * 7) return;
  int mt = wid / 7, nt = wid % 7;
  v8f acc = {};
  acc = wmma_gemm16(lane, 2 * HIDDEN,
      [&](int m, int k) { return output_enc[(mt * 16 + m) * 2 * HIDDEN + k]; },
      [&](int k, int n) {
        int f = nt * 16 + n;
        return f < ATTN_F ? attn_W[f * 769 + k] : 0.f;
      }, acc);
  wmma_store16(lane, acc, [&](int m, int n, float v) {
    int f = nt * 16 + n;
    if (f < ATTN_F) P[(mt * 16 + m) * ATTN_F + f] = v + attn_b[f];
  });
}

// ---- K4: sequential decoder phase A (attention/coverage/context/pgen/v1) ---
__global__ __launch_bounds__(512) void k_phaseA(
    const float* __restrict__ output_enc, const float* __restrict__ att_mask,
    const float* __restrict__ attn_W, const float* __restrict__ attn_out_W,
    const float* __restrict__ v1_W, const float* __restrict__ v1_b,
    const float* __restrict__ pgen_W, const float* __restrict__ pgen_b,
    const float* __restrict__ xemb, const float* __restrict__ odec,
    const float* __restrict__ P, float* __restrict__ pv1_out,
    float* __restrict__ pgen_out, float* __restrict__ aw_out,
    float* __restrict__ covloss_out, float* __restrict__ cov_out) {
  __shared__ float s_cov[L_ENC * BATCH];
  __shared__ float s_ea[L_ENC * BATCH];
  __shared__ float s_dec[BATCH * HIDDEN];
  __shared__ float s_ctx[BATCH * 2 * HIDDEN];
  __shared__ float s_dp[BATCH * 112];
  __shared__ float s_wout[112];
  __shared__ float s_wcov[112];
  __shared__ float s_pgen[BATCH];
  int tid = threadIdx.x, lane = tid & 31, wave = tid >> 5;
  for (int i = tid; i < L_ENC * BATCH; i += 512) s_cov[i] = 0.f;
  if (tid < 112) {
    s_wout[tid] = tid < ATTN_F ? attn_out_W[tid] : 0.f;
    s_wcov[tid] = tid < ATTN_F ? attn_W[tid * 769 + 768] : 0.f;
  }
  __syncthreads();
  for (int t = 0; t < T_DEC; ++t) {
    for (int i = tid; i < BATCH * HIDDEN; i += 512) s_dec[i] = odec[t * BATCH * HIDDEN + i];
    __syncthreads();
    if (wave < 7) {  // dec_part = dec @ Wdec.T  [16,100]
      int nt = wave;
      v8f acc = {};
      acc = wmma_gemm16(lane, HIDDEN,
          [&](int m, int k) { return s_dec[m * HIDDEN + k]; },
          [&](int k, int n) {
            int f = nt * 16 + n;
            return f < ATTN_F ? attn_W[f * 769 + 2 * HIDDEN + k] : 0.f;
          }, acc);
      wmma_store16(lane, acc,
                   [&](int m, int n, float v) { s_dp[m * 112 + nt * 16 + n] = v; });
    }
    __syncthreads();
    // attention energies
    for (int i = tid; i < L_ENC * BATCH; i += 512) {
      int b = i & 15;
      float cv = s_cov[i];
      const float* Pr = P + i * ATTN_F;
      float e = 0.f;
      for (int f = 0; f < ATTN_F; ++f)
        e += s_wout[f] * tanhf(Pr[f] + s_dp[b * 112 + f] + cv * s_wcov[f]);
      s_ea[i] = e + att_mask[i];
    }
    __syncthreads();
    // per-column softmax over L, coverage loss, coverage update (one wave/column)
    if (wave < BATCH) {
      int b = wave;
      float mx = -3.0e38f;
      for (int l = lane; l < L_ENC; l += 32) mx = fmaxf(mx, s_ea[l * 16 + b]);
      for (int o = 16; o; o >>= 1) mx = fmaxf(mx, __shfl_xor(mx, o));
      float sm = 0.f;
      for (int l = lane; l < L_ENC; l += 32) {
        float ex = __expf(s_ea[l * 16 + b] - mx);
        s_ea[l * 16 + b] = ex;
        sm += ex;
      }
      for (int o = 16; o; o >>= 1) sm += __shfl_xor(sm, o);
      float inv = 1.f / sm;
      float cl = 0.f;
      for (int l = lane; l < L_ENC; l += 32) {
        int ix = l * 16 + b;
        float a = s_ea[ix] * inv;
        s_ea[ix] = a;
        float cv = s_cov[ix];
        cl += fminf(a, cv);
        s_cov[ix] = cv + a;
      }
      for (int o = 16; o; o >>= 1) cl += __shfl_xor(cl, o);
      if (lane == 0) covloss_out[t * BATCH + b] = cl;
    }
    __syncthreads();
    // context = sum_l a * output_enc
    for (int i = tid; i < BATCH * 2 * HIDDEN; i += 512) {
      int b = i >> 9, h = i & 511;
      float s = 0.f;
      for (int l = 0; l < L_ENC; ++l)
        s += s_ea[l * 16 + b] * output_enc[(l * 16 + b) * 2 * HIDDEN + h];
      s_ctx[i] = s;
    }
    __syncthreads();
    // pgen
    if (tid < BATCH) {
      int b = tid;
      float s = pgen_b[0];
      for (int k = 0; k < 2 * HIDDEN; ++k) s += s_ctx[b * 2 * HIDDEN + k] * pgen_W[k];
      for (int k = 0; k < HIDDEN; ++k) s += s_dec[b * HIDDEN + k] * pgen_W[2 * HIDDEN + k];
      const float* emb = xemb + t * BATCH * GLOVE + b * GLOVE;
      for (int k = 0; k < GLOVE; ++k) s += emb[k] * pgen_W[3 * HIDDEN + k];
      float pg = sigf(s);
      s_pgen[b] = pg;
      pgen_out[t * BATCH + b] = pg;
    }
    __syncthreads();
    // copy-mass weights
    for (int i = tid; i < L_ENC * BATCH; i += 512)
      aw_out[t * L_ENC * BATCH + i] = (1.f - s_pgen[i & 15]) * s_ea[i];
    // pv1 = [dec, ctx] @ v1_W.T + v1_b
    if (wave < 7) {
      int nt = wave;
      v8f acc = {};
      acc = wmma_gemm16(lane, 3 * HIDDEN,
          [&](int m, int k) {
            return k < HIDDEN ? s_dec[m * HIDDEN + k] : s_ctx[m * 2 * HIDDEN + (k - HIDDEN)];
          },
          [&](int k, int n) {
            int f = nt * 16 + n;
            return f < VOCAB_F ? v1_W[f * 3 * HIDDEN + k] : 0.f;
          }, acc);
      wmma_store16(lane, acc, [&](int m, int n, float v) {
        int f = nt * 16 + n;
        if (f < VOCAB_F) pv1_out[t * BATCH * VOCAB_F + m * VOCAB_F + f] = v + v1_b[f];
      });
    }
    __syncthreads();
  }
  for (int i = tid; i < L_ENC * BATCH; i += 512) cov_out[i] = s_cov[i];
}

// ---- K5a: logits[t] = pv1[t] @ v2_W.T + v2_b  (WMMA, grid over t,vtile) ----
__global__ __launch_bounds__(256) void k_vocab_logits(
    const float* __restrict__ pv1, const float* __restrict__ v2_W,
    const float* __restrict__ v2_b, float* __restrict__ out) {
  int tid = threadIdx.x, lane = tid & 31, wave = tid >> 5;
  int vt = blockIdx.x * 8 + wave;
  int t = blockIdx.y;
  if (vt >= VOCAB / 16) return;
  const float* A = pv1 + t * BATCH * VOCAB_F;
  v8f acc = {};
  acc = wmma_gemm16(lane, VOCAB_F,
      [&](int m, int k) { return A[m * VOCAB_F + k]; },
      [&](int k, int n) { return v2_W[(vt * 16 + n) * VOCAB_F + k]; }, acc);
  wmma_store16(lane, acc, [&](int m, int n, float v) {
    int vcol = vt * 16 + n;
    out[(size_t)t * BATCH * BIG_V + (size_t)m * BIG_V + vcol] = v + v2_b[vcol];
  });
}

// ---- K5b: per-(t,b) row max & sumexp over 50000 logits ---------------------
__global__ __launch_bounds__(256) void k_rowstat(const float* __restrict__ out,
                                                 float* __restrict__ rmax,
                                                 float* __restrict__ rsum) {
  int row = blockIdx.x;  // t*16+b
  const float* p = out + (size_t)row * BIG_V;
  __shared__ float red[256];
  float mx = -3.0e38f;
  for (int v = threadIdx.x; v < VOCAB; v += 256) mx = fmaxf(mx, p[v]);
  red[threadIdx.x] = mx;
  __syncthreads();
  for (int s = 128; s > 0; s >>= 1) {
    if (threadIdx.x < s) red[threadIdx.x] = fmaxf(red[threadIdx.x], red[threadIdx.x + s]);
    __syncthreads();
  }
  mx = red[0];
  __syncthreads();
  float sm = 0.f;
  for (int v = threadIdx.x; v < VOCAB; v += 256) sm += __expf(p[v] - mx);
  red[threadIdx.x] = sm;
  __syncthreads();
  for (int s = 128; s > 0; s >>= 1) {
    if (threadIdx.x < s) red[threadIdx.x] += red[threadIdx.x + s];
    __syncthreads();
  }
  if (threadIdx.x == 0) { rmax[row] = mx; rsum[row] = red[0]; }
}

// ---- K5c: logits -> pgen * softmax, zero the extended-vocab pad ------------
__global__ void k_pv(float* __restrict__ out, const float* __restrict__ rmax,
                     const float* __restrict__ rsum, const float* __restrict__ pgen) {
  size_t n = (size_t)T_DEC * BATCH * BIG_V;
  for (size_t i = (size_t)blockIdx.x * blockDim.x + threadIdx.x; i < n;
       i += (size_t)gridDim.x * blockDim.x) {
    int row = (int)(i / BIG_V);
    int v = (int)(i % BIG_V);
    float val = 0.f;
    if (v < VOCAB) val = __expf(out[i] - rmax[row]) / rsum[row] * pgen[row];
    out[i] = val;
  }
}

// ---- K6: copy-mass scatter; one thread per (t,b) row -> deterministic ------
__global__ void k_scatter(float* __restrict__ out, const float* __restrict__ aw,
                          const int* __restrict__ real_index) {
  int i = blockIdx.x * blockDim.x + threadIdx.x;
  if (i >= T_DEC * BATCH) return;
  int t = i >> 4, b = i & 15;
  float* row = out + (size_t)i * BIG_V;
  for (int l = 0; l < L_ENC; ++l)
    row[real_index[l * BATCH + b]] += aw[t * L_ENC * BATCH + l * BATCH + b];
}

// ---- K7: final log ---------------------------------------------------------
__global__ void k_log(float* __restrict__ out) {
  size_t n = (size_t)T_DEC * BATCH * BIG_V;
  for (size_t i = (size_t)blockIdx.x * blockDim.x + threadIdx.x; i < n;
       i += (size_t)gridDim.x * blockDim.x)
    out[i] = logf(out[i]);
}

extern "C" void kernel_launch(void* const* d_in, const int* in_sizes, int n_in,
                              void* d_out, int out_size, void* d_ws, size_t ws_size,
                              hipStream_t stream) {
  (void)in_sizes; (void)n_in; (void)out_size; (void)ws_size;
  const float* output_enc = (const float*)d_in[0];
  const float* hidden_enc = (const float*)d_in[1];
  const float* cell_enc   = (const float*)d_in[2];
  const float* att_mask   = (const float*)d_in[3];
  const float* embed_tab  = (const float*)d_in[4];
  const float* W_ih       = (const float*)d_in[5];
  const float* W_hh       = (const float*)d_in[6];
  const float* b_ih       = (const float*)d_in[7];
  const float* b_hh       = (const float*)d_in[8];
  const float* reduce_W   = (const float*)d_in[9];
  const float* reduce_b   = (const float*)d_in[10];
  const float* attn_W     = (const float*)d_in[11];
  const float* attn_b     = (const float*)d_in[12];
  const float* attn_out_W = (const float*)d_in[13];
  const float* v1_W       = (const float*)d_in[14];
  const float* v1_b       = (const float*)d_in[15];
  const float* v2_W       = (const float*)d_in[16];
  const float* v2_b       = (const float*)d_in[17];
  const float* pgen_W     = (const float*)d_in[18];
  const float* pgen_b     = (const float*)d_in[19];
  const int*   real_index = (const int*)d_in[20];
  const int*   input_dec  = (const int*)d_in[21];
  float* ws  = (float*)d_ws;
  float* out = (float*)d_out;

  k_reduce_init<<<1, 1024, 0, stream>>>(hidden_enc, cell_enc, reduce_W, reduce_b,
                                        ws + WS_H0, ws + WS_C0);
  k_embed<<<(T_DEC * BATCH * GLOVE + 255) / 256, 256, 0, stream>>>(embed_tab, input_dec,
                                                                   ws + WS_XEMB);
  k_lstm<<<1, 512, 0, stream>>>(W_ih, W_hh, b_ih, b_hh, ws + WS_H0, ws + WS_C0,
                                ws + WS_XEMB, ws + WS_ODEC, out + OUT_HT, out + OUT_CT);
  k_attn_pre<<<350, 256, 0, stream>>>(output_enc, attn_W, attn_b, ws + WS_P);
  k_phaseA<<<1, 512, 0, stream>>>(output_enc, att_mask, attn_W, attn_out_W, v1_W, v1_b,
                                  pgen_W, pgen_b, ws + WS_XEMB, ws + WS_ODEC, ws + WS_P,
                                  ws + WS_PV1, ws + WS_PGEN, ws + WS_AW, out + OUT_CL,
                                  out + OUT_COV);
  k_vocab_logits<<<dim3(391, T_DEC), 256, 0, stream>>>(ws + WS_PV1, v2_W, v2_b, out);
  k_rowstat<<<T_DEC * BATCH, 256, 0, stream>>>(out, ws + WS_RMAX, ws + WS_RSUM);
  k_pv<<<4096, 256, 0, stream>>>(out, ws + WS_RMAX, ws + WS_RSUM, ws + WS_PGEN);
  k_scatter<<<4, 256, 0, stream>>>(out, ws + WS_AW, real_index);
  k_log<<<4096, 256, 0, stream>>>(out);
}